// BURE_65154653880840
// MI455X (gfx1250) — compile-verified
//
#include <hip/hip_runtime.h>
#include <math.h>

// ---------------- model dims ----------------
#define BB     2
#define TT     16
#define DD     256
#define HW     1024          // H*W = 32*32
#define NSEQ   2048          // B*H*W
#define MROWS  32768         // NSEQ*T
#define DIN    512           // D_INNER
#define DST    16            // D_STATE
#define DTR    16            // DT_RANK
#define NCLS   9
#define NH     8
#define HD     32
#define COUT1  128           // D/2

typedef __attribute__((ext_vector_type(2))) float        v2f;
typedef __attribute__((ext_vector_type(8))) float        v8f;
typedef __attribute__((ext_vector_type(4))) unsigned int v4u;
typedef __attribute__((ext_vector_type(4))) int          v4i;
typedef __attribute__((ext_vector_type(8))) int          v8i;

#if defined(__has_builtin)
#  if __has_builtin(__builtin_amdgcn_tensor_load_to_lds)
#    define HAVE_TDM 1
#  endif
#endif
#ifndef HAVE_TDM
#  define HAVE_TDM 0
#endif
// amdgpu-toolchain (clang-23/therock) ships this header and uses the 6-arg
// builtin; ROCm 7.2 (clang-22) has the 5-arg form.
#if __has_include(<hip/amd_detail/amd_gfx1250_TDM.h>)
#  define TDM_SIX_ARGS 1
#else
#  define TDM_SIX_ARGS 0
#endif

__device__ __forceinline__ float siluf(float x)     { return x / (1.f + __expf(-x)); }
__device__ __forceinline__ float softplusf(float x) { return (x > 20.f) ? x : log1pf(__expf(x)); }

// ======================================================================
// C[M,N] = A[M,K] @ W[N,K]^T (+bias) (+softplus) via fp32 WMMA.
//
// Block = 8 waves, all sharing one 16-column tile tn; wave w computes the
// 16x16 C tile (tmbase + w, tn). The 16xK weight tile (<=32KB, K<=512) is
// staged into LDS ONCE per block by the Tensor Data Mover
// (TENSOR_LOAD_TO_LDS + s_wait_tensorcnt), then all 8 waves read B
// fragments from LDS. Requires ldw == K (true for every weight here) and
// M % 128 == 0 (M = 32768 for all calls).
//
// NOTE: the TDM intrinsic's LDS write is invisible to LLVM alias analysis;
// without an escape, loads of Wt get folded to undef (observed in round-2
// disasm: wmma src1 aliased the accumulator, no ds_load emitted). The
// asm-volatile memory clobber below pins Wt as "may be written".
//
// Fragment layouts per CDNA5 ISA 7.12.2:
//   A  : lane L, reg j -> A[M = L%16][K = k0 + j + 2*(L/16)]
//   B  : lane L, reg j -> W[N = L%16][K = k0 + j + 2*(L/16)]   (B = W^T)
//   C/D: lane L, reg j -> C[M = j + 8*(L/16)][N = L%16]
// ======================================================================
__global__ void k_gemm_wmma(const float* __restrict__ A, int lda,
                            const float* __restrict__ W, int ldw,
                            const float* __restrict__ bias,
                            float* __restrict__ C, int ldc,
                            int M, int N, int K, int act)
{
    __shared__ float Wt[16 * DIN];          // sole LDS allocation -> offset 0

    int lane   = threadIdx.x & 31;
    int waveId = threadIdx.x >> 5;
    int tiles_n = N >> 4;
    int tn     = blockIdx.x % tiles_n;
    int tm     = (blockIdx.x / tiles_n) * 8 + waveId;

    // ---- stage W[tn*16 .. tn*16+15][0..K-1] into LDS ----
#if HAVE_TDM
    if (threadIdx.x < 32) {                 // wave 0 only (wave-uniform)
        unsigned long long ga =
            (unsigned long long)(const void*)(W + (size_t)(tn * 16) * ldw);
        v4u g0;
        g0[0] = 1u;                                           // count=1
        g0[1] = 0u;                                           // lds_addr = 0
        g0[2] = (unsigned)(ga & 0xFFFFFFFFull);               // global_addr lo
        g0[3] = (unsigned)((ga >> 32) & 0x01FFFFFFull)        // global_addr hi
              | (2u << 30);                                   // type = 2 (image)
        v8i g1;
        g1[0] = (int)(2u << 16);                              // data_size = 4B
        g1[1] = (int)(((unsigned)K & 0xFFFFu) << 16);         // tensor_dim0 lo
        g1[2] = (int)(((unsigned)K >> 16) | (16u << 16));     // td0 hi | tensor_dim1 lo
        g1[3] = (int)(((unsigned)K & 0xFFFFu) << 16);         // td1 hi(0) | tile_dim0=K
        g1[4] = 16;                                           // tile_dim1=16, tile_dim2=0
        g1[5] = K;                                            // tensor_dim0_stride lo
        g1[6] = 0;                                            // stride0 hi | stride1 lo
        g1[7] = 0;
        v4i z4 = {};
#  if TDM_SIX_ARGS
        v8i z8 = {};
        __builtin_amdgcn_tensor_load_to_lds(g0, g1, z4, z4, z8, 0);
#  else
        __builtin_amdgcn_tensor_load_to_lds(g0, g1, z4, z4, 0);
#  endif
        __builtin_amdgcn_s_wait_tensorcnt(0);
    }
#else
    for (int i = threadIdx.x; i < 16 * K; i += 256)
        Wt[i] = W[(size_t)(tn * 16 + i / K) * ldw + (i % K)];
#endif
    __syncthreads();
    // Escape Wt's address so the compiler must assume the TDM wrote it;
    // prevents folding of the LDS loads below to undef.
    asm volatile("" : : "r"(&Wt[0]) : "memory");

    int row = lane & 15;        // M index for A frag, N index for B/C frags
    int sel = lane >> 4;        // K half-select

    const float* Ap = A + (size_t)(tm * 16 + row) * lda + 2 * sel;
    const float* Bp = Wt + row * K + 2 * sel;
    __builtin_prefetch(Ap, 0, 1);           // global_prefetch_b8 (A rows)

    v8f acc = {};
    for (int k0 = 0; k0 < K; k0 += 4) {
        v2f a, b;
        a.x = Ap[k0];     a.y = Ap[k0 + 1];
        b.x = Bp[k0];     b.y = Bp[k0 + 1];
        acc = __builtin_amdgcn_wmma_f32_16x16x4_f32(
            false, a, false, b, (short)0, acc, false, false);
    }

    float bv = bias ? bias[tn * 16 + row] : 0.f;
    #pragma unroll
    for (int j = 0; j < 8; ++j) {
        int mrow = tm * 16 + j + 8 * sel;
        float v = acc[j] + bv;
        if (act == 1) v = softplusf(v);
        C[(size_t)mrow * ldc + tn * 16 + row] = v;
    }
}

// ---------------- E (B,T,D,H,W) -> E_flat (MROWS, D) ----------------
__global__ void k_transposeE(const float* __restrict__ E, float* __restrict__ ef)
{
    int idx = blockIdx.x * blockDim.x + threadIdx.x;
    if (idx >= MROWS * DD) return;
    int d = idx % DD;
    int m = idx / DD;
    int t = m % TT;
    int n = m / TT;
    int p = n % HW;
    int b = n / HW;
    ef[idx] = E[((size_t)(b * TT + t) * DD + d) * HW + p];
}

// -------- depthwise causal conv1d (D_CONV=4) + SiLU; xin = xz[:, :512] --------
__global__ void k_conv1d_silu(const float* __restrict__ xz,
                              const float* __restrict__ cw,
                              const float* __restrict__ cb,
                              float* __restrict__ u)
{
    int idx = blockIdx.x * blockDim.x + threadIdx.x;
    if (idx >= MROWS * DIN) return;
    int c = idx % DIN;
    int m = idx / DIN;
    int t = m % TT;
    int n = m / TT;
    float acc = cb[c];
    #pragma unroll
    for (int j = 0; j < 4; ++j) {
        int tt = t - 3 + j;
        if (tt >= 0) acc += xz[(size_t)(n * TT + tt) * 1024 + c] * cw[c * 4 + j];
    }
    u[idx] = siluf(acc);
}

// -------- selective scan: one block (512 thr) per sequence n --------
// yg[m,d] = ( scan_y + D*u ) * silu(z);   yg may alias u (read-before-write).
__global__ void k_scan(const float* __restrict__ u,  const float* __restrict__ dt,
                       const float* __restrict__ xdbl, const float* __restrict__ xz,
                       const float* __restrict__ A_log, const float* __restrict__ Dp,
                       float* __restrict__ yg)
{
    __shared__ float Bs[TT][DST];
    __shared__ float Cs[TT][DST];
    int n = blockIdx.x;
    int tid = threadIdx.x;            // 0..511
    if (tid < 256) {
        int t = tid >> 4, s = tid & 15;
        Bs[t][s] = xdbl[(size_t)(n * TT + t) * 48 + DTR + s];
    } else {
        int q = tid - 256;
        int t = q >> 4, s = q & 15;
        Cs[t][s] = xdbl[(size_t)(n * TT + t) * 48 + DTR + DST + s];
    }
    __syncthreads();

    int d = tid;
    float nA[DST], h[DST];
    #pragma unroll
    for (int s = 0; s < DST; ++s) {
        nA[s] = -__expf(A_log[d * DST + s]);
        h[s]  = 0.f;
    }
    float Dv = Dp[d];
    for (int t = 0; t < TT; ++t) {
        size_t m = (size_t)n * TT + t;
        float dtv = dt[m * DIN + d];
        float uv  = u [m * DIN + d];
        float du  = dtv * uv;
        float y = 0.f;
        #pragma unroll
        for (int s = 0; s < DST; ++s) {
            h[s] = __expf(dtv * nA[s]) * h[s] + du * Bs[t][s];
            y += h[s] * Cs[t][s];
        }
        y += Dv * uv;
        float z = xz[m * 1024 + DIN + d];
        yg[m * DIN + d] = y * siluf(z);
    }
}

// -------- tiny attention: 1 thread per (n, head, query) --------
__global__ void k_attn(const float* __restrict__ q, const float* __restrict__ k,
                       const float* __restrict__ v, float* __restrict__ o)
{
    int idx = blockIdx.x * blockDim.x + threadIdx.x;
    if (idx >= NSEQ * NH * TT) return;
    int qi = idx % TT;
    int h  = (idx / TT) % NH;
    int n  = idx / (TT * NH);
    const float scale = 0.17677669529663687f;   // 1/sqrt(32)

    float qv[HD];
    const float* qp = q + ((size_t)(n * TT + qi) * DD) + h * HD;
    #pragma unroll
    for (int j = 0; j < HD; ++j) qv[j] = qp[j];

    float sc[TT]; float mx = -1e30f;
    #pragma unroll
    for (int ki = 0; ki < TT; ++ki) {
        const float* kp = k + ((size_t)(n * TT + ki) * DD) + h * HD;
        float s = 0.f;
        #pragma unroll
        for (int j = 0; j < HD; ++j) s += qv[j] * kp[j];
        s *= scale; sc[ki] = s; mx = fmaxf(mx, s);
    }
    float den = 0.f;
    #pragma unroll
    for (int ki = 0; ki < TT; ++ki) { sc[ki] = __expf(sc[ki] - mx); den += sc[ki]; }
    float inv = 1.f / den;

    float out[HD];
    #pragma unroll
    for (int j = 0; j < HD; ++j) out[j] = 0.f;
    #pragma unroll
    for (int ki = 0; ki < TT; ++ki) {
        const float* vp = v + ((size_t)(n * TT + ki) * DD) + h * HD;
        float pw = sc[ki] * inv;
        #pragma unroll
        for (int j = 0; j < HD; ++j) out[j] += pw * vp[j];
    }
    float* op = o + ((size_t)(n * TT + qi) * DD) + h * HD;
    #pragma unroll
    for (int j = 0; j < HD; ++j) op[j] = out[j];
}

// -------- residual + LayerNorm, writes C in (B,T,D,H,W) layout to d_out --------
__global__ void k_ln(const float* __restrict__ a, const float* __restrict__ ef,
                     const float* __restrict__ g, const float* __restrict__ be,
                     float* __restrict__ out)
{
    __shared__ float sb[256];
    __shared__ float stat[2];
    int m = blockIdx.x, d = threadIdx.x;
    float x = a[(size_t)m * DD + d] + ef[(size_t)m * DD + d];
    sb[d] = x; __syncthreads();
    for (int s = 128; s > 0; s >>= 1) { if (d < s) sb[d] += sb[d + s]; __syncthreads(); }
    if (d == 0) stat[0] = sb[0] * (1.f / DD);
    __syncthreads();
    float mu = stat[0];
    float dx = x - mu;
    sb[d] = dx * dx; __syncthreads();
    for (int s = 128; s > 0; s >>= 1) { if (d < s) sb[d] += sb[d + s]; __syncthreads(); }
    if (d == 0) stat[1] = rsqrtf(sb[0] * (1.f / DD) + 1e-5f);
    __syncthreads();
    float xn = dx * stat[1] * g[d] + be[d];
    int t = m % TT, n = m / TT;
    int p = n % HW, b = n / HW;
    out[((size_t)(b * TT + t) * DD + d) * HW + p] = xn;
}

// -------- prev_feat / curr_feat means over C --------
__global__ void k_feat(const float* __restrict__ C,
                       float* __restrict__ prevf, float* __restrict__ currf)
{
    __shared__ float sb[256];
    int bd = blockIdx.x;            // b*256 + d
    int b = bd >> 8, d = bd & 255;
    int tid = threadIdx.x;

    float s = 0.f;
    for (int i = tid; i < 15 * HW; i += 256) {
        int t = i / HW, p = i % HW;
        s += C[((size_t)(b * TT + t) * DD + d) * HW + p];
    }
    sb[tid] = s; __syncthreads();
    for (int st = 128; st > 0; st >>= 1) { if (tid < st) sb[tid] += sb[tid + st]; __syncthreads(); }
    if (tid == 0) prevf[bd] = sb[0] / (15.f * HW);
    __syncthreads();

    float s2 = 0.f;
    for (int p = tid; p < HW; p += 256)
        s2 += C[((size_t)(b * TT + 15) * DD + d) * HW + p];
    sb[tid] = s2; __syncthreads();
    for (int st = 128; st > 0; st >>= 1) { if (tid < st) sb[tid] += sb[tid + st]; __syncthreads(); }
    if (tid == 0) currf[bd] = sb[0] / (float)HW;
}

// -------- temporal-consistency head + prior loss (tiny, one block) --------
__global__ void k_head(const float* __restrict__ prevf, const float* __restrict__ currf,
                       const float* __restrict__ tw, const float* __restrict__ tb,
                       float* __restrict__ pl, float* __restrict__ ev, float* __restrict__ prior)
{
    int i = threadIdx.x;    // 32 threads
    if (i < BB * NCLS) {
        int b = i / NCLS, c = i % NCLS;
        float s1 = 0.f, s2 = 0.f;
        for (int d = 0; d < DD; ++d) {
            float w = tw[c * DD + d];
            s1 += prevf[b * DD + d] * w;
            s2 += currf[b * DD + d] * w;
        }
        pl[i] = s1 + tb[c];
        ev[i] = softplusf(s2 + tb[c]);
    }
    __syncthreads();
    if (i == 0) {
        float S[NCLS];
        for (int c = 0; c < NCLS; ++c) {
            float s = 0.f;
            for (int b = 0; b < BB; ++b) s += __expf(pl[b * NCLS + c]) + 1.f;
            S[c] = s / BB;
        }
        float tot = 0.f;
        for (int b = 0; b < BB; ++b) {
            float mx = -1e30f;
            for (int c = 0; c < NCLS; ++c) mx = fmaxf(mx, pl[b * NCLS + c]);
            float yn[NCLS]; float den = 0.f;
            for (int c = 0; c < NCLS; ++c) { yn[c] = __expf(pl[b * NCLS + c] - mx); den += yn[c]; }
            float row = 0.f;
            for (int c = 0; c < NCLS; ++c)
                row += (yn[c] / den) * (logf(S[c] + 1e-8f) - logf(ev[b * NCLS + c] + 1.f));
            tot += row;
        }
        prior[0] = tot / BB;
    }
}

// -------- 3x3 SAME conv on C_curr: (B,256,32,32) -> (B,128,32,32) --------
__global__ void k_conv3x3(const float* __restrict__ C, const float* __restrict__ w,
                          const float* __restrict__ bias, float* __restrict__ h1)
{
    int idx = blockIdx.x * blockDim.x + threadIdx.x;
    if (idx >= BB * COUT1 * HW) return;
    int x  = idx & 31;
    int y  = (idx >> 5) & 31;
    int oc = (idx >> 10) & (COUT1 - 1);
    int b  = idx >> 17;
    float acc = bias[oc];
    for (int ic = 0; ic < DD; ++ic) {
        const float* inp = C + ((size_t)(b * TT + 15) * DD + ic) * HW;
        const float* wp  = w + ((size_t)oc * DD + ic) * 9;
        #pragma unroll
        for (int ky = 0; ky < 3; ++ky) {
            int yy = y + ky - 1;
            if ((unsigned)yy >= 32u) continue;
            #pragma unroll
            for (int kx = 0; kx < 3; ++kx) {
                int xx = x + kx - 1;
                if ((unsigned)xx >= 32u) continue;
                acc += inp[yy * 32 + xx] * wp[ky * 3 + kx];
            }
        }
    }
    h1[idx] = acc;
}

// -------- BN stats per channel (mean, population var) --------
__global__ void k_bnstats(const float* __restrict__ h1, float* __restrict__ bm, float* __restrict__ bv)
{
    __shared__ float s1[256], s2[256];
    int oc = blockIdx.x, tid = threadIdx.x;
    float a = 0.f, b2 = 0.f;
    for (int i = tid; i < BB * HW; i += 256) {
        int b = i / HW, p = i % HW;
        float v = h1[(size_t)(b * COUT1 + oc) * HW + p];
        a += v; b2 += v * v;
    }
    s1[tid] = a; s2[tid] = b2; __syncthreads();
    for (int st = 128; st > 0; st >>= 1) {
        if (tid < st) { s1[tid] += s1[tid + st]; s2[tid] += s2[tid + st]; }
        __syncthreads();
    }
    if (tid == 0) {
        float mu = s1[0] / (float)(BB * HW);
        bm[oc] = mu;
        bv[oc] = s2[0] / (float)(BB * HW) - mu * mu;
    }
}

// accumulator layout in scalar area
#define ACC_FOCAL 0
#define ACC_CE    1
#define ACC_INT   2            // [2,9]
#define ACC_PCS   20           // [2,9]
#define ACC_TCS   38           // [2,9]
#define ACC_NZERO 56

__global__ void k_zero(float* p, int n)
{
    int i = blockIdx.x * blockDim.x + threadIdx.x;
    if (i < n) p[i] = 0.f;
}

// -------- per-pixel BN+ReLU+1x1conv+log_softmax+CE/focal/dice accumulation --------
__global__ void k_seg(const float* __restrict__ h1, const float* __restrict__ bm,
                      const float* __restrict__ bv, const float* __restrict__ bg,
                      const float* __restrict__ bb, const float* __restrict__ c2w,
                      const float* __restrict__ c2b, const int* __restrict__ labels,
                      float* __restrict__ acc)
{
    int idx = blockIdx.x * blockDim.x + threadIdx.x;
    if (idx >= BB * HW) return;
    int b = idx / HW, p = idx % HW;

    float pred[NCLS];
    #pragma unroll
    for (int c = 0; c < NCLS; ++c) pred[c] = c2b[c];
    for (int oc = 0; oc < COUT1; ++oc) {
        float v  = h1[(size_t)(b * COUT1 + oc) * HW + p];
        float hn = (v - bm[oc]) * rsqrtf(bv[oc] + 1e-5f) * bg[oc] + bb[oc];
        hn = fmaxf(hn, 0.f);
        #pragma unroll
        for (int c = 0; c < NCLS; ++c) pred[c] += hn * c2w[c * COUT1 + oc];
    }
    float mx = -1e30f;
    #pragma unroll
    for (int c = 0; c < NCLS; ++c) mx = fmaxf(mx, pred[c]);
    float den = 0.f;
    #pragma unroll
    for (int c = 0; c < NCLS; ++c) den += __expf(pred[c] - mx);
    float lse = mx + logf(den);

    int lab = labels[idx];
    float ce = lse - pred[lab];
    float pt = __expf(-ce);
    float om = 1.f - pt;
    atomicAdd(&acc[ACC_FOCAL], om * om * ce);
    atomicAdd(&acc[ACC_CE], ce);
    #pragma unroll
    for (int c = 0; c < NCLS; ++c) {
        float psm = __expf(pred[c] - lse);
        atomicAdd(&acc[ACC_PCS + b * NCLS + c], psm);
        if (c == lab) {
            atomicAdd(&acc[ACC_INT + b * NCLS + c], psm);
            atomicAdd(&acc[ACC_TCS + b * NCLS + c], 1.f);
        }
    }
}

// -------- final scalar loss --------
__global__ void k_final(const float* __restrict__ acc, const float* __restrict__ prior,
                        const float* __restrict__ ltheta, float* __restrict__ out)
{
    if (threadIdx.x != 0 || blockIdx.x != 0) return;
    const float inv_np = 1.f / (float)(BB * HW);
    float focal = acc[ACC_FOCAL] * inv_np;
    float dce   = acc[ACC_CE]    * inv_np;
    float dl = 0.f;
    for (int c = 0; c < NCLS; ++c) {
        float mb = 0.f;
        for (int b = 0; b < BB; ++b) {
            float inter = acc[ACC_INT + b * NCLS + c];
            float un    = acc[ACC_PCS + b * NCLS + c] + acc[ACC_TCS + b * NCLS + c];
            mb += (2.f * inter + 1e-6f) / (un + 1e-6f);
        }
        dl += 1.f - mb / BB;
    }
    dl /= (float)NCLS;
    float losses[3] = { focal, dl, dce };
    float post = 0.f;
    for (int i = 0; i < 3; ++i) {
        float th = 1.f / (1.f + __expf(-ltheta[i])) + 1e-8f;
        post += losses[i] / (2.f * th * th) + logf(1.f + th * th);
    }
    out[(size_t)MROWS * DD] = prior[0] + post;   // element 8388608
}

// ======================================================================
// host side
// ======================================================================
static inline void launch_gemm(const float* A, int lda, const float* W, int ldw,
                               const float* bias, float* C, int ldc,
                               int M, int N, int K, int act, hipStream_t s)
{
    // requires: M % 128 == 0, N % 16 == 0, K % 4 == 0, K <= DIN, ldw == K
    int tiles_m = M / 16;
    int tiles_n = N / 16;
    int blocks  = (tiles_m / 8) * tiles_n;
    k_gemm_wmma<<<blocks, 256, 0, s>>>(A, lda, W, ldw, bias, C, ldc, M, N, K, act);
}

extern "C" void kernel_launch(void* const* d_in, const int* in_sizes, int n_in,
                              void* d_out, int out_size, void* d_ws, size_t ws_size,
                              hipStream_t stream)
{
    const float* E        = (const float*)d_in[0];
    const int*   labels   = (const int*)  d_in[1];
    const float* in_proj  = (const float*)d_in[2];
    const float* conv_w   = (const float*)d_in[3];
    const float* conv_b   = (const float*)d_in[4];
    const float* x_proj   = (const float*)d_in[5];
    const float* dt_w     = (const float*)d_in[6];
    const float* dt_b     = (const float*)d_in[7];
    const float* A_log    = (const float*)d_in[8];
    const float* D_param  = (const float*)d_in[9];
    const float* out_proj = (const float*)d_in[10];
    const float* q_w = (const float*)d_in[11]; const float* q_b = (const float*)d_in[12];
    const float* k_w = (const float*)d_in[13]; const float* k_b = (const float*)d_in[14];
    const float* v_w = (const float*)d_in[15]; const float* v_b = (const float*)d_in[16];
    const float* o_w = (const float*)d_in[17]; const float* o_b = (const float*)d_in[18];
    const float* ln_g = (const float*)d_in[19]; const float* ln_b = (const float*)d_in[20];
    const float* tc_w = (const float*)d_in[21]; const float* tc_b = (const float*)d_in[22];
    const float* c1_w = (const float*)d_in[23]; const float* c1_b = (const float*)d_in[24];
    const float* bn_g = (const float*)d_in[25]; const float* bn_b = (const float*)d_in[26];
    const float* c2_w = (const float*)d_in[27]; const float* c2_b = (const float*)d_in[28];
    const float* log_theta = (const float*)d_in[29];

    float* out = (float*)d_out;           // C (8388608 floats) + loss (1 float)
    float* ws  = (float*)d_ws;

    // ---- workspace layout (floats), with aliasing ----
    float* ef   = ws;                          // 8388608   E_flat
    float* xz   = ef + (size_t)8388608;        // 33554432  in_proj out (xin|z)
    float* u    = xz + (size_t)33554432;       // 16777216  conv+silu out; ygate alias; 'a' alias
    float* xd   = u  + (size_t)16777216;       // 1572864   x_dbl; h1 alias
    float* dt   = xd + (size_t)1572864;        // 16777216  dt; scalar area alias
    float* mbuf = dt + (size_t)16777216;       // 8388608   mamba output
    // aliases (regions dead by the time aliases are used):
    float* qb   = xz;
    float* kb   = xz + (size_t)8388608;
    float* vb   = xz + (size_t)16777216;
    float* attn = xz + (size_t)25165824;
    float* abuf = u;
    float* h1   = xd;
    float* prevf = dt;            // 512
    float* currf = dt + 512;      // 512
    float* pl    = dt + 1024;     // 18
    float* ev    = dt + 1042;     // 18
    float* prior = dt + 1060;     // 1
    float* accum = dt + 1100;     // ACC_NZERO floats
    float* bm    = dt + 1200;     // 128
    float* bv    = dt + 1328;     // 128

    // ---- Mamba branch ----
    k_transposeE<<<(MROWS * DD + 255) / 256, 256, 0, stream>>>(E, ef);

    launch_gemm(ef, DD, in_proj, DD, nullptr, xz, 2 * DIN, MROWS, 2 * DIN, DD, 0, stream);

    k_conv1d_silu<<<(MROWS * DIN + 255) / 256, 256, 0, stream>>>(xz, conv_w, conv_b, u);

    launch_gemm(u, DIN, x_proj, DIN, nullptr, xd, 48, MROWS, 48, DIN, 0, stream);

    // dt = softplus(x_dbl[:, :16] @ dt_w^T + dt_b)   (A stride 48, K=16)
    launch_gemm(xd, 48, dt_w, DTR, dt_b, dt, DIN, MROWS, DIN, DTR, 1, stream);

    k_scan<<<NSEQ, DIN, 0, stream>>>(u, dt, xd, xz, A_log, D_param, u /*ygate alias*/);

    launch_gemm(u, DIN, out_proj, DIN, nullptr, mbuf, DD, MROWS, DD, DIN, 0, stream);

    // ---- cross attention ----
    launch_gemm(ef,   DD, q_w, DD, q_b, qb, DD, MROWS, DD, DD, 0, stream);
    launch_gemm(mbuf, DD, k_w, DD, k_b, kb, DD, MROWS, DD, DD, 0, stream);
    launch_gemm(mbuf, DD, v_w, DD, v_b, vb, DD, MROWS, DD, DD, 0, stream);

    k_attn<<<(NSEQ * NH * TT + 255) / 256, 256, 0, stream>>>(qb, kb, vb, attn);

    launch_gemm(attn, DD, o_w, DD, o_b, abuf, DD, MROWS, DD, DD, 0, stream);

    // ---- residual + LayerNorm -> C (d_out) ----
    k_ln<<<MROWS, 256, 0, stream>>>(abuf, ef, ln_g, ln_b, out);

    // ---- heads / losses ----
    k_zero<<<1, 64, 0, stream>>>(accum, ACC_NZERO);

    k_feat<<<BB * DD, 256, 0, stream>>>(out, prevf, currf);
    k_head<<<1, 32, 0, stream>>>(prevf, currf, tc_w, tc_b, pl, ev, prior);

    k_conv3x3<<<(BB * COUT1 * HW + 255) / 256, 256, 0, stream>>>(out, c1_w, c1_b, h1);
    k_bnstats<<<COUT1, 256, 0, stream>>>(h1, bm, bv);
    k_seg<<<(BB * HW + 255) / 256, 256, 0, stream>>>(h1, bm, bv, bn_g, bn_b,
                                                     c2_w, c2_b, labels, accum);
    k_final<<<1, 1, 0, stream>>>(accum, prior, log_theta, out);

    (void)in_sizes; (void)n_in; (void)out_size; (void)ws_size;
}